// TextureMartingaleModule_2199023256203
// MI455X (gfx1250) — compile-verified
//
#include <hip/hip_runtime.h>
#include <hip/hip_bf16.h>

// TextureMartingale: 3x3 zero-padded stencil -> 4 texture features -> clamp
// transform (THETA=1 martingale reduces exactly to clamp(f*e^-0.5, 1e-4, 1e4)).
//
// Data path: gfx1250 async global->LDS halo staging (ASYNCcnt) + LDS reuse.
// Zero padding is materialized IN LDS during staging, so the per-pixel inner
// loop is branchless: 6x ds_load_b96 + straight-line (dual-issuable) math.
// Math path: sum-identity variance (no sqrt), staged per-element entropy
// terms (1.13 logs/pixel instead of 9), fast rcp divisions.
// Staging address math: single integer division per thread; incremental
// (hy,hx) updates; 'ok' flags held in registers across the ASYNCcnt wait.

#define TILE  32
#define HALO  34             // TILE + 2 (3x3 stencil halo)
#define HELEMS (HALO * HALO) // 1156
#define PLANE_W 512
#define PLANE_ELEMS (512 * 512)

#define EXPM05 (0.60653066f)      // e^-0.5
#define INV9   (0.11111111f)

__global__ void __launch_bounds__(256)
texture_martingale_kernel(const float* __restrict__ x, float* __restrict__ out)
{
    __shared__ float tile[HELEMS];   // zero-padded halo tile
    __shared__ float tlog[HELEMS];   // staged cl*log(cl) per (padded) element

    const int plane = blockIdx.z;            // b*4 + c, 0..31
    const int tX = blockIdx.x * TILE;        // tile origin x
    const int tY = blockIdx.y * TILE;        // tile origin y
    const int tid = threadIdx.x;             // 0..255

    // Uniform per-block base address of this (b,c) plane -> SGPR pair.
    const unsigned long long xbase =
        (unsigned long long)x + (unsigned long long)plane * (PLANE_ELEMS * 4ull);

    // ---- Stage 34x34 halo into LDS via gfx1250 async global->LDS ----
    // Source coords clamped to the plane (pure OOB guard); pass 2 replaces
    // out-of-plane elements with true zero padding.
    int hy = tid / HALO;                 // the only integer division
    int hx = tid - hy * HALO;
    const unsigned lbase   = (unsigned)(unsigned long long)(&tile[0]) + 4u * (unsigned)tid;
    const unsigned lds1155 = (unsigned)(unsigned long long)(&tile[HELEMS - 1]);
    bool okf[5];

#pragma unroll
    for (int i = 0; i < 5; ++i) {
        const int  idx   = tid + i * 256;
        const bool valid = (i < 4) || (idx < HELEMS);
        // Tail lanes alias element 1155 (same data, same address as its owner).
        const int ehy = valid ? hy : (HALO - 1);
        const int ehx = valid ? hx : (HALO - 1);
        const int gy  = tY + ehy - 1;
        const int gx  = tX + ehx - 1;
        okf[i] = ((unsigned)gy < 512u) && ((unsigned)gx < 512u);

        int cy = gy < 0 ? 0 : (gy > 511 ? 511 : gy);
        int cx = gx < 0 ? 0 : (gx > 511 ? 511 : gx);
        const unsigned goff = ((unsigned)((cy << 9) + cx)) << 2;       // bytes
        const unsigned lds_byte = valid ? (lbase + (unsigned)i * 1024u) : lds1155;

        asm volatile("global_load_async_to_lds_b32 %0, %1, %2"
                     :
                     : "v"(lds_byte), "v"(goff), "s"(xbase)
                     : "memory");

        // advance (hy,hx) by 256 elements: 256 = 7*34 + 18
        hx += 18;
        const bool c = (hx >= HALO);
        hy += c ? 8 : 7;
        hx -= c ? HALO : 0;
    }

#if __has_builtin(__builtin_amdgcn_s_wait_asynccnt)
    __builtin_amdgcn_s_wait_asynccnt(0);
#else
    asm volatile("s_wait_asynccnt 0" ::: "memory");
#endif
    __syncthreads();   // tile[] raw data complete for all waves

    // ---- Materialize zero padding + entropy terms, once per element ----
    // Each idx owned by one thread; writes race-free before the barrier.
    // Addresses are base + immediate offsets (idx = tid + i*256).
#pragma unroll
    for (int i = 0; i < 5; ++i) {
        const int idx = tid + i * 256;
        if (idx < HELEMS) {
            const float v = okf[i] ? tile[idx] : 0.0f;
            if (!okf[i]) tile[idx] = 0.0f;        // true zero padding in LDS
            const float cl = fmaxf(v, 1e-6f);     // v==0 -> cl*log(cl) == pad term
            tlog[idx] = cl * __logf(cl);
        }
    }
    __syncthreads();

    // ---- Compute: each thread handles 4 pixels (rows strided by 8) ----
    const int tx = tid & (TILE - 1);
    const int ty = tid >> 5;                 // 0..7
    const int gx = tX + tx;

#pragma unroll
    for (int k = 0; k < 4; ++k) {
        const int r  = ty + (k << 3);        // tile row 0..31
        const int gy = tY + r;

        const float* pr0 = &tile[(r + 0) * HALO + tx];
        const float* pr1 = &tile[(r + 1) * HALO + tx];
        const float* pr2 = &tile[(r + 2) * HALO + tx];
        const float* tr0 = &tlog[(r + 0) * HALO + tx];
        const float* tr1 = &tlog[(r + 1) * HALO + tx];
        const float* tr2 = &tlog[(r + 2) * HALO + tx];

        float p[9];
        p[0] = pr0[0]; p[1] = pr0[1]; p[2] = pr0[2];
        p[3] = pr1[0]; p[4] = pr1[1]; p[5] = pr1[2];
        p[6] = pr2[0]; p[7] = pr2[1]; p[8] = pr2[2];

        const float ST = ((tr0[0] + tr0[1]) + (tr0[2] + tr1[0]))
                       + ((tr1[1] + tr1[2]) + (tr2[0] + tr2[1])) + tr2[2];

        float S = 0.0f, E2 = 0.0f;
#pragma unroll
        for (int i = 0; i < 9; ++i) {
            S  += p[i];
            E2  = fmaf(p[i], p[i], E2);
        }

        const float mean = S * INV9;
        float ad = 0.0f;
#pragma unroll
        for (int i = 0; i < 9; ++i) {
            ad += fabsf(p[i] - mean);
        }

        // m2 = sum((p-mean)^2) = E2 - S^2/9 ; clip(sqrt(var),1e-3)^2 == max(var,1e-6)
        const float m2  = fmaf(-mean, S, E2);
        const float var = m2 * 0.125f;                  // Bessel /(K-1)
        const float den = fmaxf(var, 1e-6f);

        const float contrast = __fdividef(m2, den) * INV9;
        const float energy   = E2 * INV9;
        const float entropy  = ST * (-INV9);
        const float homog    = __fdividef(1.0f, fmaf(ad, INV9, 1.0f) + 1e-6f);

        const float f4[4] = {contrast, energy, entropy, homog};

        // out[b, c*4 + f, gy, gx]  with (b*16 + c*4) == plane*4
        float* outp = out + (unsigned long long)(plane * 4) * PLANE_ELEMS
                          + (unsigned)(gy * PLANE_W + gx);
#pragma unroll
        for (int f = 0; f < 4; ++f) {
            // THETA=1 martingale == clamp(f * e^-0.5, 1e-4, 1e4) exactly
            float M = EXPM05 * fmaxf(f4[f], 1e-5f);
            M = fminf(fmaxf(M, 1e-4f), 1e4f);       // folds to a med3-style clamp
            outp[(unsigned long long)f * PLANE_ELEMS] = M;
        }
    }
}

extern "C" void kernel_launch(void* const* d_in, const int* in_sizes, int n_in,
                              void* d_out, int out_size, void* d_ws, size_t ws_size,
                              hipStream_t stream)
{
    (void)in_sizes; (void)n_in; (void)out_size; (void)d_ws; (void)ws_size;
    const float* x = (const float*)d_in[0];
    float* out = (float*)d_out;

    // grid: 16x16 tiles per 512x512 plane, 32 planes (B*C)
    dim3 grid(16, 16, 32);
    dim3 block(256);
    hipLaunchKernelGGL(texture_martingale_kernel, grid, block, 0, stream, x, out);
}